// MacridTEARSLLoss_48515950576386
// MI455X (gfx1250) — compile-verified
//
#include <hip/hip_runtime.h>

#define INFF __builtin_inff()
#define BLK 512

// ---------- helpers ----------

__device__ __forceinline__ void lse_merge(float& m, float& s, float m2, float s2) {
  float M = fmaxf(m, m2);
  s = s * __expf(m - M) + s2 * __expf(m2 - M);
  m = M;
}

__device__ __forceinline__ void online4(float& m, float& s, const float4& v) {
  float c4 = fmaxf(fmaxf(v.x, v.y), fmaxf(v.z, v.w));
  float M  = fmaxf(m, c4);
  s = s * __expf(m - M) + __expf(v.x - M) + __expf(v.y - M)
                        + __expf(v.z - M) + __expf(v.w - M);
  m = M;
}

__device__ __forceinline__ float dot4(float d, const float4& a, const float4& b) {
  d = fmaf(a.x, b.x, d); d = fmaf(a.y, b.y, d);
  d = fmaf(a.z, b.z, d); d = fmaf(a.w, b.w, d);
  return d;
}

__device__ __forceinline__ float wave_sum(float v) {
#pragma unroll
  for (int o = 16; o > 0; o >>= 1) v += __shfl_down(v, o, 32);
  return v;
}

__device__ __forceinline__ void wave_lse(float& m, float& s) {
#pragma unroll
  for (int o = 16; o > 0; o >>= 1) {
    float m2 = __shfl_down(m, o, 32);
    float s2 = __shfl_down(s, o, 32);
    lse_merge(m, s, m2, s2);
  }
}

// CDNA5 async global -> LDS copy (per-lane b128), tracked by ASYNCcnt.
__device__ __forceinline__ void async_ld_b128(unsigned lds_off, unsigned long long gaddr) {
  asm volatile("global_load_async_to_lds_b128 %0, %1, off"
               :: "v"(lds_off), "v"(gaddr) : "memory");
}

// ---------- kernel 1: one block per row ----------

__global__ __launch_bounds__(BLK) void row_kernel(
    const float* __restrict__ recon, const float* __restrict__ x,
    const float* __restrict__ mu,    const float* __restrict__ logvar,
    const float* __restrict__ ltext, const float* __restrict__ lrec,
    const float* __restrict__ pmu,   const float* __restrict__ plv,
    float* __restrict__ ws, int N, int D)
{
  const int row  = blockIdx.x;
  const int tid  = threadIdx.x;
  const int nf4  = N >> 2;             // float4 elements per row
  const int full = nf4 / BLK;          // full 512-wide chunks
  const int tail = nf4 - full * BLK;

  __shared__ float4 xbuf[2][BLK];      // double-buffered x tile (async target)
  __shared__ float  red[BLK / 32][13];

  const float4* r0 = (const float4*)(recon + (size_t)row * N);
  const float4* r1 = (const float4*)(ltext + (size_t)row * N);
  const float4* r2 = (const float4*)(lrec  + (size_t)row * N);
  const float4* xr = (const float4*)(x     + (size_t)row * N);

  float m0 = -INFF, s0 = 0.f, d0 = 0.f;
  float m1 = -INFF, s1 = 0.f, d1 = 0.f;
  float m2 = -INFF, s2 = 0.f, d2 = 0.f;
  float xs = 0.f;

  // prologue: async-copy chunk 0 of x into LDS buffer 0
  if (full > 0) {
    unsigned lo = (unsigned)(unsigned long long)&xbuf[0][tid]; // low 32 bits == LDS offset
    async_ld_b128(lo, (unsigned long long)(xr + tid));
  }

  for (int c = 0; c < full; ++c) {
    const int i = c * BLK + tid;
    float4 v0 = r0[i];
    float4 v1 = r1[i];
    float4 v2 = r2[i];
    if (c + 1 < full) {
      // our ds reads of the buffer being overwritten finished one iter ago;
      // fence them before the async store can land
      asm volatile("s_wait_dscnt 0x0" ::: "memory");
      unsigned lo = (unsigned)(unsigned long long)&xbuf[(c + 1) & 1][tid];
      async_ld_b128(lo, (unsigned long long)(xr + (c + 1) * BLK + tid));
      __builtin_prefetch(r0 + (c + 1) * BLK + tid, 0, 3);
      __builtin_prefetch(r1 + (c + 1) * BLK + tid, 0, 3);
      __builtin_prefetch(r2 + (c + 1) * BLK + tid, 0, 3);
      asm volatile("s_wait_asynccnt 0x1" ::: "memory"); // chunk c arrived, c+1 in flight
    } else {
      asm volatile("s_wait_asynccnt 0x0" ::: "memory");
    }
    float4 xv = xbuf[c & 1][tid];
    online4(m0, s0, v0); d0 = dot4(d0, v0, xv);
    online4(m1, s1, v1); d1 = dot4(d1, v1, xv);
    online4(m2, s2, v2); d2 = dot4(d2, v2, xv);
    xs += (xv.x + xv.y) + (xv.z + xv.w);
  }

  if (tid < tail) {                    // tail chunk: direct loads
    const int i = full * BLK + tid;
    float4 v0 = r0[i], v1 = r1[i], v2 = r2[i], xv = xr[i];
    online4(m0, s0, v0); d0 = dot4(d0, v0, xv);
    online4(m1, s1, v1); d1 = dot4(d1, v1, xv);
    online4(m2, s2, v2); d2 = dot4(d2, v2, xv);
    xs += (xv.x + xv.y) + (xv.z + xv.w);
  }

  // latent terms for this row (D = 400)
  float k1 = 0.f, k2 = 0.f, w = 0.f;
  for (int d = tid; d < D; d += BLK) {
    size_t o = (size_t)row * D + d;
    float lv = logvar[o], pl = plv[o];
    float e1 = __expf(lv), e2 = __expf(pl);
    float dm = mu[o] - pmu[o];
    k1 += 1.f + lv - e1;
    k2 += 1.f + pl - e2;
    w  += dm * dm + e1 + e2 - 2.f * sqrtf(e1 * e2);   // 2*tr(sqrt(S1 S2)) for diagonals
  }

  // wave-level reduction
  wave_lse(m0, s0); wave_lse(m1, s1); wave_lse(m2, s2);
  d0 = wave_sum(d0); d1 = wave_sum(d1); d2 = wave_sum(d2);
  xs = wave_sum(xs); k1 = wave_sum(k1); k2 = wave_sum(k2); w = wave_sum(w);

  const int wv = tid >> 5, ln = tid & 31;
  if (ln == 0) {
    float* r = red[wv];
    r[0]=m0; r[1]=s0; r[2]=d0; r[3]=m1; r[4]=s1; r[5]=d1;
    r[6]=m2; r[7]=s2; r[8]=d2; r[9]=xs; r[10]=k1; r[11]=k2; r[12]=w;
  }
  __syncthreads();

  if (tid < 32) {
    const int NW = BLK / 32;
    bool ok = tid < NW;
    const float* r = red[ok ? tid : 0];
    m0 = ok ? r[0] : -INFF; s0 = ok ? r[1] : 0.f; d0 = ok ? r[2]  : 0.f;
    m1 = ok ? r[3] : -INFF; s1 = ok ? r[4] : 0.f; d1 = ok ? r[5]  : 0.f;
    m2 = ok ? r[6] : -INFF; s2 = ok ? r[7] : 0.f; d2 = ok ? r[8]  : 0.f;
    xs = ok ? r[9] : 0.f;   k1 = ok ? r[10] : 0.f;
    k2 = ok ? r[11] : 0.f;  w  = ok ? r[12] : 0.f;
    wave_lse(m0, s0); wave_lse(m1, s1); wave_lse(m2, s2);
    d0 = wave_sum(d0); d1 = wave_sum(d1); d2 = wave_sum(d2);
    xs = wave_sum(xs); k1 = wave_sum(k1); k2 = wave_sum(k2); w = wave_sum(w);
    if (tid == 0) {
      float lse0 = m0 + logf(s0);
      float lse1 = m1 + logf(s1);
      float lse2 = m2 + logf(s2);
      float* o = ws + (size_t)row * 6;
      o[0] = d0 - lse0 * xs;   // Σ_n (recon - lse)·x  (merged)
      o[1] = d1 - lse1 * xs;   // text
      o[2] = d2 - lse2 * xs;   // rec
      o[3] = k1; o[4] = k2; o[5] = w;
    }
  }
}

// ---------- kernel 2: combine the 256 per-row partials ----------

__global__ __launch_bounds__(256) void final_kernel(
    const float* __restrict__ ws, float* __restrict__ out, int B, int N, int D)
{
  const int tid = threadIdx.x;
  float v[6] = {0.f, 0.f, 0.f, 0.f, 0.f, 0.f};
  for (int b = tid; b < B; b += 256) {
    const float* r = ws + (size_t)b * 6;
#pragma unroll
    for (int j = 0; j < 6; ++j) v[j] += r[j];
  }
#pragma unroll
  for (int j = 0; j < 6; ++j) v[j] = wave_sum(v[j]);

  __shared__ float red[8][6];
  const int wv = tid >> 5, ln = tid & 31;
  if (ln == 0) {
#pragma unroll
    for (int j = 0; j < 6; ++j) red[wv][j] = v[j];
  }
  __syncthreads();
  if (tid < 32) {
#pragma unroll
    for (int j = 0; j < 6; ++j) v[j] = (tid < 8) ? red[tid][j] : 0.f;
#pragma unroll
    for (int j = 0; j < 6; ++j) v[j] = wave_sum(v[j]);
    if (tid == 0) {
      const float invBN = 1.f / ((float)B * (float)N);
      const float invBD = 1.f / ((float)B * (float)D);
      float bm = -v[0] * invBN;            // BCE_merged
      float bt = -v[1] * invBN;            // BCE_text
      float br = -v[2] * invBN;            // BCE_rec
      float BCE = (bm + bt + br) * (1.f / 3.f);
      float K1 = -0.5f * v[3] * invBD;
      float K2 = -0.5f * v[4] * invBD;
      float W  = v[5] / (float)B;
      float l  = BCE + 0.5f * (K1 + K2) + W;   // anneal = epsilon = 1
      out[0] = l; out[1] = BCE; out[2] = W;
      out[3] = br; out[4] = bt; out[5] = bm;
    }
  }
}

extern "C" void kernel_launch(void* const* d_in, const int* in_sizes, int n_in,
                              void* d_out, int out_size, void* d_ws, size_t ws_size,
                              hipStream_t stream) {
  const float* recon = (const float*)d_in[0];
  const float* x     = (const float*)d_in[1];
  const float* mu    = (const float*)d_in[2];
  const float* lv    = (const float*)d_in[3];
  const float* lt    = (const float*)d_in[4];
  const float* lr    = (const float*)d_in[5];
  const float* pmu   = (const float*)d_in[6];
  const float* plv   = (const float*)d_in[7];
  const int D = 400;
  const int B = in_sizes[2] / D;        // 256
  const int N = in_sizes[0] / B;        // 20000
  float* ws = (float*)d_ws;

  row_kernel<<<B, BLK, 0, stream>>>(recon, x, mu, lv, lt, lr, pmu, plv, ws, N, D);
  final_kernel<<<1, 256, 0, stream>>>(ws, (float*)d_out, B, N, D);
}